// GCN_50732153701091
// MI455X (gfx1250) — compile-verified
//
#include <hip/hip_runtime.h>

#define NNODES 100000
#define F_IN 48
#define HID 128

typedef __attribute__((ext_vector_type(2))) float v2f;
typedef __attribute__((ext_vector_type(8))) float v8f;

// ---------------------------------------------------------------------------
// agg = src  (elementwise copy, float4 vectorized)
// ---------------------------------------------------------------------------
__global__ __launch_bounds__(256) void copy_f4_kernel(const float4* __restrict__ src,
                                                      float4* __restrict__ dst, int n4) {
  int i = blockIdx.x * 256 + threadIdx.x;
  if (i < n4) dst[i] = src[i];
}

// ---------------------------------------------------------------------------
// agg[dst] += feat[src]  over all edges; each thread handles one float4 chunk
// of one edge. Hardware global_atomic_add_f32 via unsafeAtomicAdd.
// ---------------------------------------------------------------------------
template <int F4>
__global__ __launch_bounds__(256) void scatter_add_kernel(const float* __restrict__ feat,
                                                          const int* __restrict__ srcIdx,
                                                          const int* __restrict__ dstIdx,
                                                          float* __restrict__ agg,
                                                          unsigned total) {
  unsigned idx = blockIdx.x * 256u + threadIdx.x;
  if (idx >= total) return;
  unsigned e = idx / F4;
  unsigned c = (idx - e * F4) * 4u;
  const int s = srcIdx[e];
  const int d = dstIdx[e];
  const float4 v = *(const float4*)(feat + (size_t)s * (F4 * 4) + c);
  float* o = agg + (size_t)d * (F4 * 4) + c;
  unsafeAtomicAdd(o + 0, v.x);
  unsafeAtomicAdd(o + 1, v.y);
  unsafeAtomicAdd(o + 2, v.z);
  unsafeAtomicAdd(o + 3, v.w);
}

// ---------------------------------------------------------------------------
// H = relu(A @ W) + H      (A: [N,128], W: [128,128], in-place residual)
// One wave => 16 rows x 128 cols = eight 16x16 f32 WMMA tiles, K in steps of 4.
// ---------------------------------------------------------------------------
__global__ __launch_bounds__(256) void gemm128_relu_res(const float* __restrict__ A,
                                                        const float* __restrict__ W,
                                                        float* __restrict__ H, int nrows) {
  const int wave = blockIdx.x * 8 + (threadIdx.x >> 5);
  const int lane = threadIdx.x & 31;
  const int m0 = wave << 4;
  if (m0 >= nrows) return;                 // wave-uniform: EXEC stays all-1s
  const int hi = lane >> 4;                // 0 or 1
  const int l15 = lane & 15;
  const int row = m0 + l15;
  const int kh = hi * 2;                   // K sub-offset per A/B fragment layout

  v8f acc[8] = {};
  const float* Ar = A + (size_t)row * HID;
  for (int k0 = 0; k0 < HID; k0 += 4) {
    const v2f a = *(const v2f*)(Ar + k0 + kh);           // K = k0+kh, k0+kh+1
    const float* Wp = W + (size_t)(k0 + kh) * HID + l15; // row-major W
#pragma unroll
    for (int t = 0; t < 8; ++t) {
      v2f b;
      b.x = Wp[t * 16];
      b.y = Wp[t * 16 + HID];
      acc[t] = __builtin_amdgcn_wmma_f32_16x16x4_f32(false, a, false, b, (short)0,
                                                     acc[t], false, false);
    }
  }
#pragma unroll
  for (int t = 0; t < 8; ++t) {
#pragma unroll
    for (int r = 0; r < 8; ++r) {
      const size_t off = (size_t)(m0 + hi * 8 + r) * HID + t * 16 + l15;
      H[off] = fmaxf(acc[t][r], 0.0f) + H[off];
    }
  }
}

// ---------------------------------------------------------------------------
// Block 1: H = relu(AGG @ W1) + X @ Wdim1     (K = 48, two fused GEMMs)
// ---------------------------------------------------------------------------
__global__ __launch_bounds__(256) void gemm_block1(const float* __restrict__ AGG,
                                                   const float* __restrict__ X,
                                                   const float* __restrict__ W1,
                                                   const float* __restrict__ Wd,
                                                   float* __restrict__ H, int nrows) {
  const int wave = blockIdx.x * 8 + (threadIdx.x >> 5);
  const int lane = threadIdx.x & 31;
  const int m0 = wave << 4;
  if (m0 >= nrows) return;
  const int hi = lane >> 4;
  const int l15 = lane & 15;
  const int row = m0 + l15;
  const int kh = hi * 2;

  v8f acc1[8] = {};
  v8f acc2[8] = {};
  const float* Ar = AGG + (size_t)row * F_IN;
  const float* Xr = X + (size_t)row * F_IN;
  for (int k0 = 0; k0 < F_IN; k0 += 4) {
    const v2f a1 = *(const v2f*)(Ar + k0 + kh);
    const v2f a2 = *(const v2f*)(Xr + k0 + kh);
    const float* Wp1 = W1 + (size_t)(k0 + kh) * HID + l15;
    const float* Wp2 = Wd + (size_t)(k0 + kh) * HID + l15;
#pragma unroll
    for (int t = 0; t < 8; ++t) {
      v2f b1, b2;
      b1.x = Wp1[t * 16];
      b1.y = Wp1[t * 16 + HID];
      b2.x = Wp2[t * 16];
      b2.y = Wp2[t * 16 + HID];
      acc1[t] = __builtin_amdgcn_wmma_f32_16x16x4_f32(false, a1, false, b1, (short)0,
                                                      acc1[t], false, false);
      acc2[t] = __builtin_amdgcn_wmma_f32_16x16x4_f32(false, a2, false, b2, (short)0,
                                                      acc2[t], false, false);
    }
  }
#pragma unroll
  for (int t = 0; t < 8; ++t) {
#pragma unroll
    for (int r = 0; r < 8; ++r) {
      const size_t off = (size_t)(m0 + hi * 8 + r) * HID + t * 16 + l15;
      H[off] = fmaxf(acc1[t][r], 0.0f) + acc2[t][r];
    }
  }
}

// ---------------------------------------------------------------------------
extern "C" void kernel_launch(void* const* d_in, const int* in_sizes, int n_in,
                              void* d_out, int out_size, void* d_ws, size_t ws_size,
                              hipStream_t stream) {
  const float* x = (const float*)d_in[0];
  const int* ei = (const int*)d_in[1];
  // d_in[2] = edge_attr (unused by the reference forward)
  const float* Wdim1 = (const float*)d_in[3];
  const float* W1 = (const float*)d_in[4];
  const float* Wk[5] = {(const float*)d_in[5], (const float*)d_in[6], (const float*)d_in[7],
                        (const float*)d_in[8], (const float*)d_in[9]};
  float* h = (float*)d_out;     // persistent hidden state [N,128]
  float* agg = (float*)d_ws;    // conv output scratch    [N,128]

  const int N = in_sizes[0] / F_IN;
  const int E = in_sizes[1] / 2;
  const int* srcIdx = ei;       // edge_index[0]
  const int* dstIdx = ei + E;   // edge_index[1]

  const int mtiles = (N + 15) / 16;
  const int gemmBlocks = (mtiles + 7) / 8;   // 8 waves per 256-thread block

  // ---- Block 1: conv(x) -> relu(.@W1) + x@Wdim1 ----
  {
    const int n4 = N * F_IN / 4;
    copy_f4_kernel<<<(n4 + 255) / 256, 256, 0, stream>>>((const float4*)x, (float4*)agg, n4);
    const unsigned total = (unsigned)E * (F_IN / 4);
    scatter_add_kernel<F_IN / 4>
        <<<(total + 255) / 256, 256, 0, stream>>>(x, srcIdx, dstIdx, agg, total);
    gemm_block1<<<gemmBlocks, 256, 0, stream>>>(agg, x, W1, Wdim1, h, N);
  }

  // ---- Blocks 2..6: conv(h) -> relu(.@Wk) + h ----
  for (int b = 0; b < 5; ++b) {
    const int n4 = N * HID / 4;
    copy_f4_kernel<<<(n4 + 255) / 256, 256, 0, stream>>>((const float4*)h, (float4*)agg, n4);
    const unsigned total = (unsigned)E * (HID / 4);
    scatter_add_kernel<HID / 4>
        <<<(total + 255) / 256, 256, 0, stream>>>(h, srcIdx, dstIdx, agg, total);
    gemm128_relu_res<<<gemmBlocks, 256, 0, stream>>>(agg, Wk[b], h, N);
  }
}